// LengthRegulator_77360950935697
// MI455X (gfx1250) — compile-verified
//
#include <hip/hip_runtime.h>

#define B_ 32
#define T_ 512
#define D_ 256
#define K_ 3
#define R_ (D_ * K_)        // 768 = reduction length (k*D + i ordering)
#define LN_EPS 1e-5f

typedef float v2f __attribute__((ext_vector_type(2)));
typedef float v8f __attribute__((ext_vector_type(8)));

// ---------------------------------------------------------------------------
// Repack conv weights (O, I, K) -> pair-interleaved B layout:
//   wp[((r>>1)*D + n)*2 + (r&1)] = w[n, i, k],  r = k*D + i
// A WMMA B-fragment per lane is then ONE aligned 8-byte load
// (rows r0+2*khalf and r0+2*khalf+1 of column n are adjacent).
// Weights (768 KB) stay L2-resident across all 1024 M-tiles.
// ---------------------------------------------------------------------------
__global__ void repack_w_kernel(const float* __restrict__ w, float* __restrict__ wp) {
    int idx = blockIdx.x * blockDim.x + threadIdx.x;   // over R_*D_
    if (idx >= R_ * D_) return;
    int n = idx % D_;          // output channel o
    int r = idx / D_;          // r = k*D + i
    int k = r / D_;
    int i = r % D_;
    wp[(((r >> 1) * D_ + n) << 1) + (r & 1)] = w[(n * D_ + i) * K_ + k];
}

// ---------------------------------------------------------------------------
// Fused conv1d(k=3, SAME) + bias + ReLU + LayerNorm for one layer.
// Block = 16 timesteps x 256 channels. 8 waves, each computing two 16x16
// output tiles with V_WMMA_F32_16X16X4_F32 over K'=768 (192 steps).
// A-fragments: ds_load_b64 from the LDS im2col slab.
// B-fragments: global_load_b64 from the pair-interleaved weight pack.
// ---------------------------------------------------------------------------
__global__ __launch_bounds__(256)
void conv_ln_kernel(const float* __restrict__ xin,
                    const float* __restrict__ wp,      // pair-interleaved B
                    const float* __restrict__ bias,
                    const float* __restrict__ gamma,
                    const float* __restrict__ beta,
                    float* __restrict__ out) {
    __shared__ float xs[18 * D_];   // im2col tile: rows t0-1 .. t0+16
    __shared__ float ys[16 * D_];   // conv output tile (post bias+relu)

    const int b  = blockIdx.x / (T_ / 16);
    const int t0 = (blockIdx.x % (T_ / 16)) * 16;
    const int tid = threadIdx.x;

    // Stage the 18x256 input slab as float4 (zero-pad outside [0,T)).
    for (int idx = tid; idx < 18 * (D_ / 4); idx += 256) {
        int row = idx >> 6;          // D_/4 == 64 float4 per row
        int c4  = idx & 63;
        int t = t0 - 1 + row;
        float4 v = (t >= 0 && t < T_)
                       ? ((const float4*)&xin[((size_t)b * T_ + t) * D_])[c4]
                       : make_float4(0.f, 0.f, 0.f, 0.f);
        ((float4*)xs)[idx] = v;
    }
    __syncthreads();

    const int wave  = tid >> 5;
    const int lane  = tid & 31;
    const int m     = lane & 15;     // A-fragment row (M) / B-fragment col (N)
    const int khalf = lane >> 4;     // selects K sub-pair per ISA layout
    const int n0    = wave * 32;     // this wave owns output cols [n0, n0+32)

    const v2f* wpv = (const v2f*)wp; // pair-interleaved: wpv[p*D + n] = {B[2p][n], B[2p+1][n]}

    v8f c0 = {};
    v8f c1 = {};

    for (int r0 = 0; r0 < R_; r0 += 4) {
        // A fragment: A[m][r] = xs[(m + k)*D + i], r = k*D + i.
        // ISA 16x4 f32 layout: VGPR0 = {K0 | K2}, VGPR1 = {K1 | K3}
        // -> per lane the two elements are LDS-adjacent: one b64 load.
        const int k  = r0 >> 8;      // chunk never straddles a 256 boundary
        const int i0 = r0 & 255;
        v2f a = *(const v2f*)&xs[(m + k) * D_ + i0 + 2 * khalf];

        // B fragments: one b64 load per 16-wide N tile.
        const int p = (r0 >> 1) + khalf;           // pair-row index
        v2f bb0 = wpv[(size_t)p * D_ + n0 + m];
        v2f bb1 = wpv[(size_t)p * D_ + n0 + 16 + m];

        c0 = __builtin_amdgcn_wmma_f32_16x16x4_f32(false, a, false, bb0,
                                                   (short)0, c0, false, false);
        c1 = __builtin_amdgcn_wmma_f32_16x16x4_f32(false, a, false, bb1,
                                                   (short)0, c1, false, false);
    }

    // Spill accumulators (bias + ReLU) into the LDS row tile.
    // C/D layout: VGPR j holds M=j (lanes 0-15) / M=j+8 (lanes 16-31), N=lane&15.
    {
        const int rowbase = (lane >= 16) ? 8 : 0;
        const int col0 = n0 + (lane & 15);
        const int col1 = col0 + 16;
        const float bi0 = bias[col0];
        const float bi1 = bias[col1];
        #pragma unroll
        for (int j = 0; j < 8; ++j) {
            int row = rowbase + j;
            float v0 = c0[j] + bi0;
            float v1 = c1[j] + bi1;
            ys[row * D_ + col0] = v0 > 0.f ? v0 : 0.f;
            ys[row * D_ + col1] = v1 > 0.f ? v1 : 0.f;
        }
    }
    __syncthreads();

    // LayerNorm per row (wave32 shuffle reduction): 8 waves x 2 rows.
    for (int rr = 0; rr < 2; ++rr) {
        const int row = wave * 2 + rr;
        float s = 0.f;
        for (int cc = lane; cc < D_; cc += 32) s += ys[row * D_ + cc];
        #pragma unroll
        for (int off = 16; off; off >>= 1) s += __shfl_xor(s, off, 32);
        const float mean = s * (1.0f / D_);

        float sq = 0.f;
        for (int cc = lane; cc < D_; cc += 32) {
            float d = ys[row * D_ + cc] - mean;
            sq += d * d;
        }
        #pragma unroll
        for (int off = 16; off; off >>= 1) sq += __shfl_xor(sq, off, 32);
        const float inv = rsqrtf(sq * (1.0f / D_) + LN_EPS);

        for (int cc = lane; cc < D_; cc += 32) {
            float v = (ys[row * D_ + cc] - mean) * inv * gamma[cc] + beta[cc];
            out[((size_t)(b * T_ + t0 + row)) * D_ + cc] = v;
        }
    }
}

// ---------------------------------------------------------------------------
// Final D->1 projection: pred[b,t] = h2[b,t,:] . wl + bl
// ---------------------------------------------------------------------------
__global__ __launch_bounds__(256)
void linear_kernel(const float* __restrict__ h, const float* __restrict__ wl,
                   const float* __restrict__ bl, float* __restrict__ pred) {
    int idx = blockIdx.x * blockDim.x + threadIdx.x;
    if (idx >= B_ * T_) return;
    const float4* hr = (const float4*)&h[(size_t)idx * D_];
    const float4* wv = (const float4*)wl;
    float acc = 0.f;
    #pragma unroll 4
    for (int j = 0; j < D_ / 4; ++j) {
        float4 a = hr[j], w = wv[j];
        acc += a.x * w.x + a.y * w.y + a.z * w.z + a.w * w.w;
    }
    pred[idx] = acc + bl[0];
}

// ---------------------------------------------------------------------------
// Per-batch inclusive cumsum of durations (ALPHA=1.0 -> dur == target int).
// ---------------------------------------------------------------------------
__global__ __launch_bounds__(256)
void scan_kernel(const int* __restrict__ dur, int* __restrict__ ends) {
    __shared__ int s0[T_];
    __shared__ int s1[T_];
    const int b = blockIdx.x;
    const int tid = threadIdx.x;
    for (int i = tid; i < T_; i += 256) s0[i] = dur[b * T_ + i];
    __syncthreads();
    int* src = s0;
    int* dst = s1;
    for (int off = 1; off < T_; off <<= 1) {
        for (int i = tid; i < T_; i += 256)
            dst[i] = src[i] + (i >= off ? src[i - off] : 0);
        __syncthreads();
        int* tmp = src; src = dst; dst = tmp;
    }
    for (int i = tid; i < T_; i += 256) ends[b * T_ + i] = src[i];
}

// ---------------------------------------------------------------------------
// Length regulation (gather form): one block per output row (b, m).
// Binary-search the segment containing m; rows past lens[b] are zero-filled.
// Writes each output element exactly once -> no separate memset pass.
// ---------------------------------------------------------------------------
__global__ __launch_bounds__(64)
void gather_kernel(const float* __restrict__ x, const int* __restrict__ ends,
                   float* __restrict__ out, int max_len) {
    const int mrow = blockIdx.x;
    const int b = blockIdx.y;
    const int* e = &ends[b * T_];
    const int lens = e[T_ - 1];
    float4* orow = (float4*)&out[((size_t)b * max_len + mrow) * D_];
    const int tid = threadIdx.x;          // 64 threads x float4 == 256 floats
    if (mrow >= lens) {
        orow[tid] = make_float4(0.f, 0.f, 0.f, 0.f);
        return;
    }
    int lo = 0, hi = T_ - 1;              // smallest t with ends[t] > mrow
    while (lo < hi) {
        int mid = (lo + hi) >> 1;
        if (e[mid] > mrow) hi = mid; else lo = mid + 1;
    }
    const float4* xrow = (const float4*)&x[((size_t)b * T_ + lo) * D_];
    orow[tid] = xrow[tid];
}

// ---------------------------------------------------------------------------
extern "C" void kernel_launch(void* const* d_in, const int* in_sizes, int n_in,
                              void* d_out, int out_size, void* d_ws, size_t ws_size,
                              hipStream_t stream) {
    const float* x    = (const float*)d_in[0];
    const int*   tdur = (const int*)d_in[1];
    const float* w1   = (const float*)d_in[2];
    const float* b1   = (const float*)d_in[3];
    const float* g1   = (const float*)d_in[4];
    const float* be1  = (const float*)d_in[5];
    const float* w2   = (const float*)d_in[6];
    const float* b2   = (const float*)d_in[7];
    const float* g2   = (const float*)d_in[8];
    const float* be2  = (const float*)d_in[9];
    const float* wl   = (const float*)d_in[10];
    const float* bl   = (const float*)d_in[11];

    float* outp = (float*)d_out;
    // out_size = B*max_len*D + B*T  ->  recover the data-dependent max_len.
    const int max_len = (out_size - B_ * T_) / (B_ * D_);
    float* pred = outp + (size_t)B_ * max_len * D_;

    char* ws = (char*)d_ws;
    float* wp1 = (float*)ws;  ws += (size_t)R_ * D_ * sizeof(float);
    float* wp2 = (float*)ws;  ws += (size_t)R_ * D_ * sizeof(float);
    float* h1  = (float*)ws;  ws += (size_t)B_ * T_ * D_ * sizeof(float);
    float* h2  = (float*)ws;  ws += (size_t)B_ * T_ * D_ * sizeof(float);
    int*   ends = (int*)ws;

    repack_w_kernel<<<(R_ * D_ + 255) / 256, 256, 0, stream>>>(w1, wp1);
    repack_w_kernel<<<(R_ * D_ + 255) / 256, 256, 0, stream>>>(w2, wp2);

    conv_ln_kernel<<<B_ * T_ / 16, 256, 0, stream>>>(x,  wp1, b1, g1, be1, h1);
    conv_ln_kernel<<<B_ * T_ / 16, 256, 0, stream>>>(h1, wp2, b2, g2, be2, h2);

    linear_kernel<<<(B_ * T_ + 255) / 256, 256, 0, stream>>>(h2, wl, bl, pred);

    scan_kernel<<<B_, 256, 0, stream>>>(tdur, ends);
    gather_kernel<<<dim3(max_len, B_), 64, 0, stream>>>(x, ends, outp, max_len);
}